// ReciprocallyGated_Cell_34978213658987
// MI455X (gfx1250) — compile-verified
//
#include <hip/hip_runtime.h>
#include <hip/hip_bf16.h>

// Problem constants (from the reference)
#define T_STEPS 256
#define BATCH   128
#define KDIM    1536   // DH * L = 512 * 3

typedef float v2f __attribute__((ext_vector_type(2)));
typedef float v8f __attribute__((ext_vector_type(8)));

// ---------------------------------------------------------------------------
// Phase 1: input projection  xproj[t*B+b, n] = sum_k x[t,b,k] * Wi[n,k] + bi[n]
// One wave per 16-row M-tile, K-loop of V_WMMA_F32_16X16X4_F32.
// B fragments come from LDS via a per-lane strided pointer: valid lanes
// (n < 2) walk their Wi row, invalid lanes sit on a 64-byte zero slot with
// stride 0, so the unused 14 N-columns multiply by zero and EXEC stays all-1s.
// ---------------------------------------------------------------------------
__global__ __launch_bounds__(256) void rg_xproj_wmma_kernel(
    const float* __restrict__ x,     // (T*B, KDIM)
    const float* __restrict__ Wi,    // (2, KDIM)
    const float* __restrict__ bi,    // (2,)
    float* __restrict__ xproj)       // (T*B, 2)
{
    // Wi staged in LDS + a 16-float zero slot for the 14 unused B columns.
    __shared__ __align__(16) float wbuf[2 * KDIM + 16];
    const int tid = threadIdx.x;
    for (int i = tid; i < 2 * KDIM; i += 256) wbuf[i] = Wi[i];
    if (tid < 16) wbuf[2 * KDIM + tid] = 0.0f;
    __syncthreads();

    const int wave = tid >> 5;               // 0..7
    const int lane = tid & 31;
    const int tile = blockIdx.x * 8 + wave;  // 16-row tile of M = T*B
    const int m    = lane & 15;              // row within tile (A) / col N (B,D)
    const int kh   = lane >> 4;              // K-half selector

    // A fragment source: row stream, lane reads K = kk + kh*2 + {0,1}
    const long   row  = (long)tile * 16 + m;
    const float* arow = x + row * KDIM + kh * 2;

    // B fragment source: B[k][n] = Wi[n][k]; lanes with n>=2 stay on zeros.
    const int    n     = m;
    const bool   bval  = (n < 2);
    const float* bp    = wbuf + (bval ? (n * KDIM + kh * 2) : (2 * KDIM));
    const int    bstep = bval ? 16 : 0;      // floats advanced per 16-K chunk

    v8f acc = {};
    for (int kk = 0; kk < KDIM; kk += 16) {
        __builtin_prefetch(arow + kk + 256, 0, 0);   // ~1KB ahead per row stream
#pragma unroll
        for (int u = 0; u < 16; u += 4) {
            v2f a = *(const v2f*)(arow + kk + u);
            v2f b = *(const v2f*)(bp + u);           // ds_load_b64, imm offset
            acc = __builtin_amdgcn_wmma_f32_16x16x4_f32(
                false, a, false, b, (short)0, acc, false, false);
        }
        bp += bstep;
    }

    // D layout: VGPR r -> row r + 8*kh, col = lane&15. Only cols 0,1 are real.
    if (bval) {
        const float bias = bi[n];
        const long r0 = (long)tile * 16 + kh * 8;
#pragma unroll
        for (int r = 0; r < 8; ++r)
            xproj[(r0 + r) * 2 + n] = acc[r] + bias;
    }
}

// ---------------------------------------------------------------------------
// Phase 2: sequential scan. One 256-thread workgroup per batch element.
// State h,c lives in registers (6 elems/thread). Per step: 4 recurrent dot
// products reduced via wave32 shuffles + small LDS pass, then gated update.
// ---------------------------------------------------------------------------
__global__ __launch_bounds__(256) void rg_scan_kernel(
    const float* __restrict__ xproj,  // (T*B, 2) : Xc, Xh (bias included)
    const float* __restrict__ h0,     // (B, KDIM)
    const float* __restrict__ c0,     // (B, KDIM)
    const float* __restrict__ Wr,     // (2, KDIM)
    const float* __restrict__ br,     // (2,)
    const float* __restrict__ Wc,     // (2, KDIM)
    const float* __restrict__ bc,     // (2,)
    float* __restrict__ outs,         // (T, B, KDIM)
    float* __restrict__ hT,           // (B, KDIM)
    float* __restrict__ cT)           // (B, KDIM)
{
    const int b   = blockIdx.x;
    const int tid = threadIdx.x;

    float h[6], c[6], wr0[6], wr1[6], wc0[6], wc1[6];
#pragma unroll
    for (int j = 0; j < 6; ++j) {
        const int k = tid + 256 * j;
        h[j]   = h0[(long)b * KDIM + k];
        c[j]   = c0[(long)b * KDIM + k];
        wr0[j] = Wr[k];        wr1[j] = Wr[KDIM + k];
        wc0[j] = Wc[k];        wc1[j] = Wc[KDIM + k];
    }
    const float br0 = br[0], br1 = br[1], bc0 = bc[0], bc1 = bc[1];

    __shared__ float red[8][4];
    __shared__ float bcast[4];
    const int wv = tid >> 5, lane = tid & 31;

    for (int t = 0; t < T_STEPS; ++t) {
        // Partial dots for Hc = h.Wr0, Hh = h.Wr1, Cc = c.Wc0, Ch = c.Wc1
        float sHc = 0.f, sHh = 0.f, sCc = 0.f, sCh = 0.f;
#pragma unroll
        for (int j = 0; j < 6; ++j) {
            sHc += h[j] * wr0[j];  sHh += h[j] * wr1[j];
            sCc += c[j] * wc0[j];  sCh += c[j] * wc1[j];
        }
        // wave32 butterfly reduction
#pragma unroll
        for (int off = 16; off > 0; off >>= 1) {
            sHc += __shfl_xor(sHc, off, 32);
            sHh += __shfl_xor(sHh, off, 32);
            sCc += __shfl_xor(sCc, off, 32);
            sCh += __shfl_xor(sCh, off, 32);
        }
        if (lane == 0) {
            red[wv][0] = sHc; red[wv][1] = sHh;
            red[wv][2] = sCc; red[wv][3] = sCh;
        }
        __syncthreads();
        if (tid < 4) {
            float s = 0.f;
#pragma unroll
            for (int w = 0; w < 8; ++w) s += red[w][tid];
            bcast[tid] = s;
        }
        __syncthreads();

        const float Hc = bcast[0] + br0, Hh = bcast[1] + br1;
        const float Cc = bcast[2] + bc0, Ch = bcast[3] + bc1;
        const float Xc = xproj[((long)t * BATCH + b) * 2 + 0];
        const float Xh = xproj[((long)t * BATCH + b) * 2 + 1];

        // 1 - sigmoid(z) = 1 / (1 + exp(z))
        const float aCh = 1.f / (1.f + __expf(Ch));
        const float aHh = 1.f / (1.f + __expf(Hh));
        const float aHc = 1.f / (1.f + __expf(Hc));
        const float aCc = 1.f / (1.f + __expf(Cc));
        const float ghc = aCh * Xh;   // broadcast term of gh
        const float gcc = aHc * Xc;   // broadcast term of gc

        float* orow = outs + ((long)t * BATCH + b) * KDIM;
#pragma unroll
        for (int j = 0; j < 6; ++j) {
            h[j] = tanhf(ghc + aHh * h[j]);
            c[j] = tanhf(gcc + aCc * c[j]);
            orow[tid + 256 * j] = h[j];   // coalesced: consecutive tid -> consecutive k
        }
        // next iteration's red[] write is ordered by the barriers above
    }

#pragma unroll
    for (int j = 0; j < 6; ++j) {
        const int k = tid + 256 * j;
        hT[(long)b * KDIM + k] = h[j];
        cT[(long)b * KDIM + k] = c[j];
    }
}

// ---------------------------------------------------------------------------
// Launch
// ---------------------------------------------------------------------------
extern "C" void kernel_launch(void* const* d_in, const int* in_sizes, int n_in,
                              void* d_out, int out_size, void* d_ws, size_t ws_size,
                              hipStream_t stream) {
    // setup_inputs order: inputs, h0, c0, Wr, br, Wc, bc, Wi, bi
    const float* x   = (const float*)d_in[0];
    const float* h0  = (const float*)d_in[1];
    const float* c0  = (const float*)d_in[2];
    const float* Wr  = (const float*)d_in[3];
    const float* br  = (const float*)d_in[4];
    const float* Wc  = (const float*)d_in[5];
    const float* bc  = (const float*)d_in[6];
    const float* Wi  = (const float*)d_in[7];
    const float* bi  = (const float*)d_in[8];

    float* out   = (float*)d_out;
    float* outs  = out;                                         // (T,B,KDIM)
    float* hT    = out + (long)T_STEPS * BATCH * KDIM;          // (B,KDIM)
    float* cT    = hT + (long)BATCH * KDIM;                     // (B,KDIM)

    float* xproj = (float*)d_ws;    // (T*B, 2) = 256 KB scratch

    // Phase 1: M = T*B = 32768 rows -> 2048 wave-tiles -> 256 blocks x 8 waves
    rg_xproj_wmma_kernel<<<256, 256, 0, stream>>>(x, Wi, bi, xproj);
    // Phase 2: one workgroup per batch element
    rg_scan_kernel<<<BATCH, 256, 0, stream>>>(xproj, h0, c0, Wr, br, Wc, bc,
                                              outs, hT, cT);
    (void)in_sizes; (void)n_in; (void)out_size; (void)ws_size;
}